// SGCN_84035330114212
// MI455X (gfx1250) — compile-verified
//
#include <hip/hip_runtime.h>
#include <math.h>
#include <float.h>

typedef __attribute__((ext_vector_type(2))) float v2f;
typedef __attribute__((ext_vector_type(8))) float v8f;

#define F_DIM 64
#define C_DIM 40
#define C_PAD 48

// ---- prep: deg=1 (self-loop), zero-padded W [48x64] and b [48] ----
__global__ void k_init_deg_padW(float* __restrict__ deg, int N,
                                const float* __restrict__ W, const float* __restrict__ b,
                                float* __restrict__ Wp, float* __restrict__ bp) {
  int i = blockIdx.x * blockDim.x + threadIdx.x;
  if (i < N) deg[i] = 1.0f;
  if (i < C_PAD * F_DIM) {
    int row = i / F_DIM;
    Wp[i] = (row < C_DIM) ? W[i] : 0.0f;
  }
  if (i < C_PAD) bp[i] = (i < C_DIM) ? b[i] : 0.0f;
}

// ---- degree accumulation over edge targets ----
__global__ void k_deg_acc(const long long* __restrict__ dstA, float* __restrict__ deg, int E) {
  int e = blockIdx.x * blockDim.x + threadIdx.x;
  if (e < E) atomicAdd(&deg[dstA[e]], 1.0f);
}

__global__ void k_dinv(const float* __restrict__ deg, float* __restrict__ dinv, int N) {
  int i = blockIdx.x * blockDim.x + threadIdx.x;
  if (i < N) {
    float d = deg[i];
    dinv[i] = (d > 0.0f) ? (1.0f / sqrtf(d)) : 0.0f;
  }
}

// ---- self-loop term initializes output: y = dinv^2 * x ----
__global__ void k_self(const float* __restrict__ xin, const float* __restrict__ dinv,
                       float* __restrict__ y, int N) {
  long long i = (long long)blockIdx.x * blockDim.x + threadIdx.x;
  long long total = (long long)N * F_DIM;
  if (i < total) {
    int node = (int)(i >> 6);
    float di = dinv[node];
    y[i] = xin[i] * di * di;
  }
}

// ---- one wave32 per edge: y[dst] += dinv[src]*dinv[dst] * x[src] ----
__global__ void k_edge(const long long* __restrict__ srcA, const long long* __restrict__ dstA,
                       const float* __restrict__ dinv, const float* __restrict__ xin,
                       float* __restrict__ y, int E) {
  int gtid = blockIdx.x * blockDim.x + threadIdx.x;
  int wave = gtid >> 5;                         // uniform within a wave
  int lane = threadIdx.x & 31;
  int nw = (gridDim.x * blockDim.x) >> 5;
  for (int e = wave; e < E; e += nw) {
    long long s = srcA[e];
    long long d = dstA[e];
    float c = dinv[s] * dinv[d];
    const v2f xv = *(const v2f*)(xin + s * F_DIM + lane * 2);  // coalesced 256B/row
    float* yr = y + d * F_DIM + lane * 2;
    atomicAdd(yr,     c * xv.x);
    atomicAdd(yr + 1, c * xv.y);
  }
}

// ---- fused logits GEMM (f32 WMMA) + log-softmax ----
// One wave computes a 16-row x 48-col (padded) tile of x2 @ Wp^T using
// V_WMMA_F32_16X16X4_F32, K=64 -> 16 steps x 3 column tiles.
// A layout (32-bit A 16x4): lanes 0-15 hold K=k0,k0+1; lanes 16-31 K=k0+2,k0+3.
// C/D layout: VGPR r -> row r (lanes 0-15) / row 8+r (lanes 16-31), col = lane&15.
__global__ void k_gemm_lsm(const float* __restrict__ x2, const float* __restrict__ Wp,
                           const float* __restrict__ bp, float* __restrict__ out, int N) {
  const int lane = threadIdx.x & 31;
  const int tile = blockIdx.x * (blockDim.x >> 5) + (threadIdx.x >> 5);
  const int row0 = tile * 16;
  if (row0 >= N) return;                        // uniform per wave: EXEC stays all-1s

  const int l15  = lane & 15;
  const int koff = (lane >> 4) << 1;            // 0 or 2

  int ar = row0 + l15; if (ar > N - 1) ar = N - 1;   // clamp keeps EXEC full
  const float* arow = x2 + (long long)ar * F_DIM + koff;
  const float* w0 = Wp + ( 0 + l15) * F_DIM + koff;
  const float* w1 = Wp + (16 + l15) * F_DIM + koff;
  const float* w2 = Wp + (32 + l15) * F_DIM + koff;

  v8f acc0 = {}, acc1 = {}, acc2 = {};
#pragma unroll
  for (int k0 = 0; k0 < F_DIM; k0 += 4) {
    v2f a  = *(const v2f*)(arow + k0);
    v2f f0 = *(const v2f*)(w0 + k0);
    v2f f1 = *(const v2f*)(w1 + k0);
    v2f f2 = *(const v2f*)(w2 + k0);
    acc0 = __builtin_amdgcn_wmma_f32_16x16x4_f32(false, a, false, f0, (short)0, acc0, false, false);
    acc1 = __builtin_amdgcn_wmma_f32_16x16x4_f32(false, a, false, f1, (short)0, acc1, false, false);
    acc2 = __builtin_amdgcn_wmma_f32_16x16x4_f32(false, a, false, f2, (short)0, acc2, false, false);
  }

  const float b0 = bp[l15], b1 = bp[16 + l15], b2 = bp[32 + l15];
  const bool v2ok = (l15 < 8);                  // cols 32..39 valid, 40..47 padding
#pragma unroll
  for (int r = 0; r < 8; ++r) {
    float m0 = acc0[r] + b0;
    float m1 = acc1[r] + b1;
    float m2 = acc2[r] + b2;
    float mx = fmaxf(m0, m1);
    mx = fmaxf(mx, v2ok ? m2 : -FLT_MAX);
#pragma unroll
    for (int off = 1; off <= 8; off <<= 1)      // xor<16 stays inside each 16-lane half
      mx = fmaxf(mx, __shfl_xor(mx, off, 32));
    float s = expf(m0 - mx) + expf(m1 - mx) + (v2ok ? expf(m2 - mx) : 0.0f);
#pragma unroll
    for (int off = 1; off <= 8; off <<= 1)
      s += __shfl_xor(s, off, 32);
    float lse = mx + logf(s);
    int row = row0 + ((lane >> 4) << 3) + r;
    if (row < N) {
      float* orow = out + (long long)row * C_DIM;
      orow[l15]      = m0 - lse;
      orow[16 + l15] = m1 - lse;
      if (v2ok) orow[32 + l15] = m2 - lse;
    }
  }
}

extern "C" void kernel_launch(void* const* d_in, const int* in_sizes, int n_in,
                              void* d_out, int out_size, void* d_ws, size_t ws_size,
                              hipStream_t stream) {
  const float*     x  = (const float*)d_in[0];       // [N,64] f32
  const long long* ei = (const long long*)d_in[1];   // [2,E] i64
  const float*     W  = (const float*)d_in[2];       // [40,64]
  const float*     b  = (const float*)d_in[3];       // [40]
  float* out = (float*)d_out;                        // [N,40]

  const int N = in_sizes[0] / F_DIM;
  const int E = in_sizes[1] / 2;
  const long long* srcA = ei;        // edge_index[0]
  const long long* dstA = ei + E;    // edge_index[1]

  // workspace layout (floats): deg[N] dinv[N] x1[N*64] x2[N*64] Wp[48*64] bp[48]
  float* ws   = (float*)d_ws;
  float* deg  = ws;
  float* dinv = deg + N;
  float* x1   = dinv + N;
  float* xp   = x1 + (long long)N * F_DIM;
  float* Wp   = xp + (long long)N * F_DIM;
  float* bpd  = Wp + C_PAD * F_DIM;

  const int t = 256;
  int initMax = (N > C_PAD * F_DIM) ? N : C_PAD * F_DIM;
  k_init_deg_padW<<<(initMax + t - 1) / t, t, 0, stream>>>(deg, N, W, b, Wp, bpd);
  k_deg_acc<<<(E + t - 1) / t, t, 0, stream>>>(dstA, deg, E);
  k_dinv<<<(N + t - 1) / t, t, 0, stream>>>(deg, dinv, N);

  long long NF = (long long)N * F_DIM;
  int selfBlocks = (int)((NF + t - 1) / t);
  int edgeBlocks = (int)(((long long)E * 32 + t - 1) / t);   // one wave32 per edge

  // hop 1: x -> x1
  k_self<<<selfBlocks, t, 0, stream>>>(x, dinv, x1, N);
  k_edge<<<edgeBlocks, t, 0, stream>>>(srcA, dstA, dinv, x, x1, E);
  // hop 2: x1 -> xp
  k_self<<<selfBlocks, t, 0, stream>>>(x1, dinv, xp, N);
  k_edge<<<edgeBlocks, t, 0, stream>>>(srcA, dstA, dinv, x1, xp, E);

  // fused logits + log_softmax (WMMA f32)
  int tiles = (N + 15) / 16;
  int wavesPerBlock = 4;
  int blocks = (tiles + wavesPerBlock - 1) / wavesPerBlock;
  k_gemm_lsm<<<blocks, wavesPerBlock * 32, 0, stream>>>(xp, Wp, bpd, out, N);
}